// TestPoint_24257975287988
// MI455X (gfx1250) — compile-verified
//
#include <hip/hip_runtime.h>
#include <stdint.h>

#define N_NODES   100000
#define N_EDGES   3200000
#define D_FEAT    64
#define D_OUT     67            // D_FEAT + 3
#define NEG_SLOPE 0.01f

// ---------------------------------------------------------------------------
// CDNA5 inline-asm helpers
// ---------------------------------------------------------------------------

// Non-returning IEEE maximumNumber float atomic (GLOBAL_ATOMIC_MAX_NUM_F32,
// VGLOBAL op 82). Device scope: output rows are hammered from many WGPs.
__device__ __forceinline__ void atomic_max_f32_dev(float* addr, float val) {
  asm volatile("global_atomic_max_num_f32 %0, %1, off scope:SCOPE_DEV"
               :: "v"(addr), "v"(val)
               : "memory");
}

// Low 32 bits of a generic pointer derived from an LDS object == LDS byte
// offset (flat-address aperture rule: the LDS address is the low 32 bits).
__device__ __forceinline__ uint32_t lds_offset_of(const void* p) {
  return (uint32_t)(uintptr_t)p;
}

// GLOBAL_LOAD_ASYNC_TO_LDS_B64: per enabled lane, copy 8 bytes from the
// lane's global address to the lane's LDS address. Tracked with ASYNCcnt.
// NT hint: the edge-index stream is single-use; don't churn L2 with it.
__device__ __forceinline__ void async_copy_b64_to_lds(uint32_t lds_off,
                                                      const void* gaddr) {
  asm volatile("global_load_async_to_lds_b64 %0, %1, off th:TH_LOAD_NT"
               :: "v"(lds_off), "v"(gaddr)
               : "memory");
}

// Wait until ASYNCcnt <= 2: with double buffering, the 2 in-flight copies for
// tile t+1 may remain outstanding while tile t's copies are guaranteed done
// (async loads return "done" in order).
__device__ __forceinline__ void wait_asynccnt_le2() {
  asm volatile("s_wait_asynccnt 0x2" ::: "memory");
}
__device__ __forceinline__ void wait_asynccnt_le0() {
  asm volatile("s_wait_asynccnt 0x0" ::: "memory");
}

// ---------------------------------------------------------------------------
// Edge scatter-max (placed first in file so the disasm snippet shows it).
// One wave32 per edge: lane l handles features l and l+32 (two coalesced
// 128B gathers of x[src]); lanes 0..2 handle pos[src]-pos[dst]. Edge index
// tiles (32 edges/wave) are double-buffered in LDS via async copies.
// ---------------------------------------------------------------------------
__global__ void __launch_bounds__(256)
pconv_edge_kernel(const float* __restrict__ x,
                  const float* __restrict__ pos,
                  const long long* __restrict__ ei,   // [2, N_EDGES] int64
                  float* __restrict__ out) {
  __shared__ long long lds_src[8][2][32];   // 8 waves/block, 2 buffers
  __shared__ long long lds_dst[8][2][32];

  const int lane = threadIdx.x & 31;
  const int w    = threadIdx.x >> 5;
  const long long nWaves = (long long)gridDim.x * 8;
  const long long wid    = (long long)blockIdx.x * 8 + w;
  const long long stride = nWaves * 32;     // edges per sweep (tiles of 32)

  long long base = wid * 32;
  if (base >= N_EDGES) return;              // N_EDGES % 32 == 0 -> full tiles

  // Prologue: stage tile 0 into buffer 0.
  async_copy_b64_to_lds(lds_offset_of(&lds_src[w][0][lane]), ei + base + lane);
  async_copy_b64_to_lds(lds_offset_of(&lds_dst[w][0][lane]),
                        ei + N_EDGES + base + lane);

  int buf = 0;
  for (; base < N_EDGES; base += stride, buf ^= 1) {
    const long long nb = base + stride;
    if (nb < N_EDGES) {
      // Stage next tile into the other buffer, then wait only for the
      // current tile (<=2 outstanding == next tile's two copies).
      async_copy_b64_to_lds(lds_offset_of(&lds_src[w][buf ^ 1][lane]),
                            ei + nb + lane);
      async_copy_b64_to_lds(lds_offset_of(&lds_dst[w][buf ^ 1][lane]),
                            ei + N_EDGES + nb + lane);
      wait_asynccnt_le2();
      // Warm GL2 two tiles ahead (global_prefetch_b8).
      const long long pb = nb + stride;
      if (pb < N_EDGES) {
        __builtin_prefetch(ei + pb + lane, 0, 1);
        __builtin_prefetch(ei + N_EDGES + pb + lane, 0, 1);
      }
    } else {
      wait_asynccnt_le0();
    }

#pragma unroll 4
    for (int j = 0; j < 32; ++j) {
      // Wave-uniform endpoints -> scalar-friendly addressing.
      int src = __builtin_amdgcn_readfirstlane((int)lds_src[w][buf][j]);
      int dst = __builtin_amdgcn_readfirstlane((int)lds_dst[w][buf][j]);

      const float* xs = x + (long long)src * D_FEAT;
      float v0 = xs[lane];          // features 0..31   (coalesced 128B)
      float v1 = xs[lane + 32];     // features 32..63  (coalesced 128B)

      float* o = out + (long long)dst * D_OUT;
      atomic_max_f32_dev(o + lane,      v0);
      atomic_max_f32_dev(o + 32 + lane, v1);
      if (lane < 3) {
        float r = pos[(long long)src * 3 + lane]
                - pos[(long long)dst * 3 + lane];
        atomic_max_f32_dev(o + 64 + lane, r);
      }
    }
  }
}

// ---------------------------------------------------------------------------
// Init: fill aggregation buffer (the output itself) with -inf, 16B stores.
// N_NODES * D_OUT = 6,700,000 is divisible by 4.
// ---------------------------------------------------------------------------
__global__ void pconv_init_kernel(float4* __restrict__ out4, int n4) {
  int i = blockIdx.x * blockDim.x + threadIdx.x;
  if (i < n4) {
    const float ninf = -__builtin_inff();
    out4[i] = make_float4(ninf, ninf, ninf, ninf);
  }
}

// ---------------------------------------------------------------------------
// Finalize: -inf (no incoming edge) -> 0 (PyG convention), then LeakyReLU.
// ---------------------------------------------------------------------------
__device__ __forceinline__ float leaky_fix(float v) {
  if (!(v > -__builtin_inff())) v = 0.0f;       // -inf (or NaN) -> 0
  return (v >= 0.0f) ? v : NEG_SLOPE * v;
}

__global__ void pconv_finalize_kernel(float4* __restrict__ out4, int n4) {
  int i = blockIdx.x * blockDim.x + threadIdx.x;
  if (i < n4) {
    float4 v = out4[i];
    v.x = leaky_fix(v.x);
    v.y = leaky_fix(v.y);
    v.z = leaky_fix(v.z);
    v.w = leaky_fix(v.w);
    out4[i] = v;
  }
}

// ---------------------------------------------------------------------------
extern "C" void kernel_launch(void* const* d_in, const int* in_sizes, int n_in,
                              void* d_out, int out_size, void* d_ws,
                              size_t ws_size, hipStream_t stream) {
  const float*     x   = (const float*)d_in[0];        // [N_NODES, 64] f32
  const float*     pos = (const float*)d_in[1];        // [N_NODES, 3]  f32
  const long long* ei  = (const long long*)d_in[2];    // [2, N_EDGES]  i64
  float*           out = (float*)d_out;                // [N_NODES, 67] f32

  const int n4 = (N_NODES * D_OUT) / 4;                // 1,675,000
  const int tb = 256;

  pconv_init_kernel<<<(n4 + tb - 1) / tb, tb, 0, stream>>>((float4*)out, n4);
  pconv_edge_kernel<<<2048, tb, 0, stream>>>(x, pos, ei, out);
  pconv_finalize_kernel<<<(n4 + tb - 1) / tb, tb, 0, stream>>>((float4*)out,
                                                               n4);
}